// Kag_24979529793743
// MI455X (gfx1250) — compile-verified
//
#include <hip/hip_runtime.h>

typedef float v2f   __attribute__((ext_vector_type(2)));
typedef float v8f   __attribute__((ext_vector_type(8)));
typedef float f32x4 __attribute__((ext_vector_type(4)));

#define WMMA_F32(acc, a, b) \
  __builtin_amdgcn_wmma_f32_16x16x4_f32(false, (a), false, (b), (short)0, (acc), false, false)

#define BB 8
#define NN 128
#define DD 512

// ---------------------------------------------------------------------------
// K1: G = W_know^T @ W_know   [D,D] x [D,D] -> [D,D]
// One wave per 16x16 output tile; K-loop of 128 f32 WMMAs.
// A[m,k] = Wk[k,m] (transposed read), B[k,n] = Wk[k,n].
// ---------------------------------------------------------------------------
__global__ __launch_bounds__(32) void gram_kernel(const float* __restrict__ Wk,
                                                  float* __restrict__ G) {
  const int lane = threadIdx.x;
  const int half = lane >> 4;     // 0: lanes 0-15 (K=0,1), 1: lanes 16-31 (K=2,3)
  const int ln   = lane & 15;
  const int c0   = blockIdx.y * 16;
  const int d0   = blockIdx.x * 16;
  v8f acc = {};
  for (int kb = 0; kb < DD; kb += 4) {
    const int k0 = kb + half * 2;
    v2f a, b;
    a.x = Wk[(k0 + 0) * DD + c0 + ln];
    a.y = Wk[(k0 + 1) * DD + c0 + ln];
    b.x = Wk[(k0 + 0) * DD + d0 + ln];
    b.y = Wk[(k0 + 1) * DD + d0 + ln];
    acc = WMMA_F32(acc, a, b);
  }
#pragma unroll
  for (int r = 0; r < 8; ++r)
    G[(c0 + r + half * 8) * DD + d0 + ln] = acc[r];
}

// ---------------------------------------------------------------------------
// K2: zi = utt @ Wk^T ; v = utt @ G ; si_lin = utt @ Ws^T
// M = B*N = 1024, N = 512, K = 512. One wave per 16x16 tile; the A tile is
// shared by three WMMA accumulation chains.
// ---------------------------------------------------------------------------
__global__ __launch_bounds__(32) void lin_kernel(const float* __restrict__ utt,
                                                 const float* __restrict__ Wk,
                                                 const float* __restrict__ Ws,
                                                 const float* __restrict__ G,
                                                 float* __restrict__ zi,
                                                 float* __restrict__ v,
                                                 float* __restrict__ si_lin) {
  const int lane = threadIdx.x;
  const int half = lane >> 4;
  const int ln   = lane & 15;
  const int m0   = blockIdx.y * 16;   // rows (B*N), 64 tiles
  const int n0   = blockIdx.x * 16;   // cols (D),   32 tiles
  v8f az = {}, av = {}, as = {};
  for (int kb = 0; kb < DD; kb += 4) {
    const int k0 = kb + half * 2;
    v2f a, b1, b2, b3;
    a.x  = utt[(m0 + ln) * DD + k0];
    a.y  = utt[(m0 + ln) * DD + k0 + 1];
    b1.x = Wk[(n0 + ln) * DD + k0];          // Wk^T
    b1.y = Wk[(n0 + ln) * DD + k0 + 1];
    b2.x = G[(k0    ) * DD + n0 + ln];       // G (symmetric)
    b2.y = G[(k0 + 1) * DD + n0 + ln];
    b3.x = Ws[(n0 + ln) * DD + k0];          // Ws^T
    b3.y = Ws[(n0 + ln) * DD + k0 + 1];
    az = WMMA_F32(az, a, b1);
    av = WMMA_F32(av, a, b2);
    as = WMMA_F32(as, a, b3);
  }
#pragma unroll
  for (int r = 0; r < 8; ++r) {
    const int row = m0 + r + half * 8;
    zi[row * DD + n0 + ln]     = az[r];
    v[row * DD + n0 + ln]      = av[r];
    si_lin[row * DD + n0 + ln] = as[r];
  }
}

// ---------------------------------------------------------------------------
// K3: deg_inv[b,i] = 1 / (sum_j seq_adj[b,i,j] + 1e-10)
// ---------------------------------------------------------------------------
__global__ void deg_kernel(const float* __restrict__ seq, float* __restrict__ deg_inv) {
  const int row = blockIdx.x * blockDim.x + threadIdx.x;
  if (row < BB * NN) {
    float s = 0.f;
    for (int j = 0; j < NN; ++j) s += seq[row * NN + j];
    deg_inv[row] = 1.0f / (s + 1e-10f);
  }
}

// ---------------------------------------------------------------------------
// K4: the bandwidth-bound kernel. One block per (b,i).
// logits[b,i,j] = dot(edge[b,i,j,:] + utt[b,j,:], v[b,i,:]) / sqrt(D)
// edge_rep (268 MB, read-once) is streamed with non-temporal b128 loads;
// v is preloaded into 4 float4 registers per lane; wave32 shuffle reduce.
// ---------------------------------------------------------------------------
__global__ __launch_bounds__(256) void edge_dot_kernel(const float* __restrict__ edge,
                                                       const float* __restrict__ utt,
                                                       const float* __restrict__ v,
                                                       float* __restrict__ logits) {
  const int b    = blockIdx.x >> 7;
  const int i    = blockIdx.x & 127;
  const int lane = threadIdx.x & 31;
  const int wave = threadIdx.x >> 5;   // 8 waves per block
  const float* vrow = v + (size_t)(b * NN + i) * DD;
  f32x4 vr[4];
#pragma unroll
  for (int p = 0; p < 4; ++p)
    vr[p] = *(const f32x4*)(vrow + p * 128 + lane * 4);
  const float* erow0 = edge + (size_t)(b * NN + i) * NN * DD;
  for (int j = wave; j < NN; j += 8) {
    const float* er = erow0 + (size_t)j * DD;
    const float* ur = utt + (size_t)(b * NN + j) * DD;
    if (j + 8 < NN)  // speculative prefetch of next row -> global_prefetch_b8
      __builtin_prefetch(erow0 + (size_t)(j + 8) * DD + lane * 16, 0, 0);
    f32x4 acc = {0.f, 0.f, 0.f, 0.f};
#pragma unroll
    for (int p = 0; p < 4; ++p) {
      f32x4 e = __builtin_nontemporal_load((const f32x4*)(er + p * 128 + lane * 4));
      f32x4 u = *(const f32x4*)(ur + p * 128 + lane * 4);
      acc += (e + u) * vr[p];
    }
    float s = acc.x + acc.y + acc.z + acc.w;
#pragma unroll
    for (int off = 16; off > 0; off >>= 1)
      s += __shfl_xor(s, off, 32);
    if (lane == 0)
      logits[(b * NN + i) * NN + j] = s * 0.044194173824159216f; // 1/sqrt(512)
  }
}

// ---------------------------------------------------------------------------
// K5: masked softmax over axis i (dim 1) for fixed (b,j); attn = softmax * bk
// ---------------------------------------------------------------------------
__global__ __launch_bounds__(128) void softmax_kernel(const float* __restrict__ logits,
                                                      const float* __restrict__ bk,
                                                      float* __restrict__ attn) {
  __shared__ float red[128];
  const int b   = blockIdx.x >> 7;
  const int j   = blockIdx.x & 127;
  const int i   = threadIdx.x;
  const int idx = (b * NN + i) * NN + j;
  const float m  = bk[idx];
  const float lg = (m > 0.f) ? logits[idx] : -1e30f;
  red[i] = lg;
  __syncthreads();
  for (int s = 64; s > 0; s >>= 1) {
    if (i < s) red[i] = fmaxf(red[i], red[i + s]);
    __syncthreads();
  }
  const float mx = red[0];
  __syncthreads();
  const float e = __expf(lg - mx);
  red[i] = e;
  __syncthreads();
  for (int s = 64; s > 0; s >>= 1) {
    if (i < s) red[i] += red[i + s];
    __syncthreads();
  }
  attn[idx] = m * (e / red[0]);
}

// ---------------------------------------------------------------------------
// K6: out[b,n,:] = selu( attn^T@zi + (seq*deg_inv)@si_lin + si_lin )
// One wave per 16x16 (n,d) tile; C seeded with si_lin; two chained WMMA
// accumulations per K-step (K = 128 over i / j).
// ---------------------------------------------------------------------------
__global__ __launch_bounds__(32) void out_kernel(const float* __restrict__ attn,
                                                 const float* __restrict__ seq,
                                                 const float* __restrict__ deg_inv,
                                                 const float* __restrict__ zi,
                                                 const float* __restrict__ si_lin,
                                                 float* __restrict__ out) {
  const int lane = threadIdx.x;
  const int half = lane >> 4;
  const int ln   = lane & 15;
  const int b    = blockIdx.z;
  const int n0   = blockIdx.y * 16;
  const int d0   = blockIdx.x * 16;
  const int rowb = b * NN;
  const float dinv = deg_inv[rowb + n0 + ln];
  v8f acc;
#pragma unroll
  for (int r = 0; r < 8; ++r)
    acc[r] = si_lin[(rowb + n0 + r + half * 8) * DD + d0 + ln];
  for (int kb = 0; kb < NN; kb += 4) {
    const int k0 = kb + half * 2;
    v2f a1, b1, a2, b2;
    a1.x = attn[(rowb + k0    ) * NN + n0 + ln];     // attn^T: A[n,i]=attn[i,n]
    a1.y = attn[(rowb + k0 + 1) * NN + n0 + ln];
    b1.x = zi[(rowb + k0    ) * DD + d0 + ln];
    b1.y = zi[(rowb + k0 + 1) * DD + d0 + ln];
    a2.x = seq[(rowb + n0 + ln) * NN + k0    ] * dinv;   // norm_adj[n,j]
    a2.y = seq[(rowb + n0 + ln) * NN + k0 + 1] * dinv;
    b2.x = si_lin[(rowb + k0    ) * DD + d0 + ln];
    b2.y = si_lin[(rowb + k0 + 1) * DD + d0 + ln];
    acc = WMMA_F32(acc, a1, b1);
    acc = WMMA_F32(acc, a2, b2);
  }
  const float kScale = 1.0507009873554805f;
  const float kAlphaScale = 1.0507009873554805f * 1.6732632423543772f;
#pragma unroll
  for (int r = 0; r < 8; ++r) {
    const float x = acc[r];
    const float y = (x > 0.f) ? kScale * x : kAlphaScale * (__expf(x) - 1.f);
    out[(rowb + n0 + r + half * 8) * DD + d0 + ln] = y;
  }
}

// ---------------------------------------------------------------------------
extern "C" void kernel_launch(void* const* d_in, const int* in_sizes, int n_in,
                              void* d_out, int out_size, void* d_ws, size_t ws_size,
                              hipStream_t stream) {
  const float* utt  = (const float*)d_in[0];   // [B,N,D]
  const float* edge = (const float*)d_in[1];   // [B,N,N,D]
  const float* bk   = (const float*)d_in[2];   // [B,N,N]
  const float* seq  = (const float*)d_in[3];   // [B,N,N]
  const float* Wk   = (const float*)d_in[4];   // [D,D]
  const float* Ws   = (const float*)d_in[5];   // [D,D]
  float* out = (float*)d_out;

  float* ws     = (float*)d_ws;
  float* G      = ws;                          // D*D            = 262144
  float* zi     = G + DD * DD;                 // B*N*D          = 524288
  float* v      = zi + BB * NN * DD;
  float* si_lin = v + BB * NN * DD;
  float* logits = si_lin + BB * NN * DD;       // B*N*N          = 131072
  float* attn   = logits + BB * NN * NN;
  float* deginv = attn + BB * NN * NN;         // B*N            = 1024

  gram_kernel<<<dim3(DD / 16, DD / 16), 32, 0, stream>>>(Wk, G);
  lin_kernel<<<dim3(DD / 16, (BB * NN) / 16), 32, 0, stream>>>(utt, Wk, Ws, G, zi, v, si_lin);
  deg_kernel<<<(BB * NN + 255) / 256, 256, 0, stream>>>(seq, deginv);
  edge_dot_kernel<<<BB * NN, 256, 0, stream>>>(edge, utt, v, logits);
  softmax_kernel<<<BB * NN, 128, 0, stream>>>(logits, bk, attn);
  out_kernel<<<dim3(DD / 16, NN / 16, BB), 32, 0, stream>>>(attn, seq, deginv, zi, si_lin, out);
}